// PyTorchFeatureGrid2D_9466107920632
// MI455X (gfx1250) — compile-verified
//
#include <hip/hip_runtime.h>
#include <cstdint>

// ---------------------------------------------------------------------------
// 2D bilinear feature-grid gather, MI455X (gfx1250).
// Memory-bound: 80 MB mandatory stream @ 23.3 TB/s (~3.4us floor); 48 MB
// feature table is L2-resident (192 MB L2), so gathers are L2 hits.
// CDNA5 paths used: global_load_async_to_lds_b128 (ASYNCcnt) double-buffered
// point staging + s_wait_asynccnt, and non-temporal b128 output stores to
// protect the table's L2 residency.
// ---------------------------------------------------------------------------

#define RES      2048
#define THREADS  256           // 8 wave32 waves per block
#define PPT      4             // points per thread per chunk
#define CHUNK    (THREADS*PPT) // 1024 points per block-iteration
#define CHUNK_B  (CHUNK*2*4)   // 8192 bytes of float2 points per buffer
#define BLOCKS   1024

typedef __attribute__((ext_vector_type(4))) float v4f;  // native vector for NT stores

__global__ __launch_bounds__(THREADS)
void feature_grid2d_kernel(const float* __restrict__ pts,
                           const float* __restrict__ fp,
                           float* __restrict__ out, int n)
{
    // double-buffered point staging: 2 x 8 KB
    __shared__ __align__(16) float smem[2 * CHUNK * 2];

    const int tid = threadIdx.x;
    // Low 32 bits of the flat LDS address == workgroup-relative LDS offset.
    const uint32_t lds_thread =
        (uint32_t)(uintptr_t)(void*)&smem[0] + (uint32_t)tid * 32u;
    const unsigned long long gbase = (unsigned long long)(uintptr_t)pts;

    const int fullChunks = n / CHUNK;
    const int cstride    = (int)gridDim.x;

    // Issue one buffer fill: 32 B per thread = 2 x b128 async copies per wave.
    auto issue = [&](int buf, int chunk) {
        uint32_t ldsa = lds_thread + (uint32_t)buf * (uint32_t)CHUNK_B;
        uint32_t goff = (uint32_t)chunk * (uint32_t)CHUNK_B + (uint32_t)tid * 32u;
        asm volatile(
            "global_load_async_to_lds_b128 %0, %1, %2\n\t"
            "global_load_async_to_lds_b128 %0, %1, %2 offset:16"
            :: "v"(ldsa), "v"(goff), "s"(gbase) : "memory");
    };

    int c0 = (int)blockIdx.x;
    if (c0 < fullChunks) issue(0, c0);

    int buf = 0;
    for (int c = c0; c < fullChunks; c += cstride, buf ^= 1) {
        const int cn = c + cstride;
        if (cn < fullChunks) {
            issue(buf ^ 1, cn);                       // prefetch next chunk
            asm volatile("s_wait_asynccnt 2" ::: "memory"); // current done
        } else {
            asm volatile("s_wait_asynccnt 0" ::: "memory");
        }

        const float* my = &smem[buf * CHUNK * 2 + tid * (PPT * 2)];
        const int p0 = c * CHUNK + tid * PPT;

        __align__(16) float rr[PPT * 3];   // 12 output floats = 3 x b128 NT stores

        #pragma unroll
        for (int k = 0; k < PPT; ++k) {
            const float x = my[2 * k + 0];
            const float y = my[2 * k + 1];
            const float sx = x * (float)(RES - 1);
            const float sy = y * (float)(RES - 1);
            int xl = (int)floorf(sx);
            int yl = (int)floorf(sy);
            xl = xl < 0 ? 0 : (xl > RES - 2 ? RES - 2 : xl);
            yl = yl < 0 ? 0 : (yl > RES - 2 ? RES - 2 : yl);
            const float tx = sx - (float)xl;
            const float ty = sy - (float)yl;

            // Two rows of 6 contiguous floats each (corner pairs are adjacent).
            const float* r0 = fp + ((size_t)yl * RES + (size_t)xl) * 3;
            const float* r1 = r0 + (size_t)RES * 3;

            #pragma unroll
            for (int ch = 0; ch < 3; ++ch) {
                const float a = r0[ch], b = r0[ch + 3];
                const float d = r1[ch], e = r1[ch + 3];
                const float clo = fmaf(tx, b - a, a);
                const float chi = fmaf(tx, e - d, d);
                rr[3 * k + ch]  = fmaf(ty, chi - clo, clo);
            }
        }

        // 48 B per thread, 16 B aligned: three NT b128 stores (stream past L2).
        v4f* dst = (v4f*)(out + (size_t)p0 * 3);
        __builtin_nontemporal_store(*(const v4f*)&rr[0], dst + 0);
        __builtin_nontemporal_store(*(const v4f*)&rr[4], dst + 1);
        __builtin_nontemporal_store(*(const v4f*)&rr[8], dst + 2);
    }

    // Tail (n not a multiple of CHUNK): plain path, block 0 only.
    if (blockIdx.x == 0) {
        for (int p = fullChunks * CHUNK + tid; p < n; p += THREADS) {
            const float x = pts[2 * p + 0];
            const float y = pts[2 * p + 1];
            const float sx = x * (float)(RES - 1);
            const float sy = y * (float)(RES - 1);
            int xl = (int)floorf(sx);
            int yl = (int)floorf(sy);
            xl = xl < 0 ? 0 : (xl > RES - 2 ? RES - 2 : xl);
            yl = yl < 0 ? 0 : (yl > RES - 2 ? RES - 2 : yl);
            const float tx = sx - (float)xl;
            const float ty = sy - (float)yl;
            const float* r0 = fp + ((size_t)yl * RES + (size_t)xl) * 3;
            const float* r1 = r0 + (size_t)RES * 3;
            for (int ch = 0; ch < 3; ++ch) {
                const float a = r0[ch], b = r0[ch + 3];
                const float d = r1[ch], e = r1[ch + 3];
                const float clo = fmaf(tx, b - a, a);
                const float chi = fmaf(tx, e - d, d);
                out[(size_t)p * 3 + ch] = fmaf(ty, chi - clo, clo);
            }
        }
    }
}

extern "C" void kernel_launch(void* const* d_in, const int* in_sizes, int n_in,
                              void* d_out, int out_size, void* d_ws, size_t ws_size,
                              hipStream_t stream) {
    const float* pts = (const float*)d_in[0];   // (N, 2) float32
    const float* fp  = (const float*)d_in[1];   // (2048, 2048, 3) float32
    float* out = (float*)d_out;                 // (N, 3) float32
    const int n = in_sizes[0] / 2;
    feature_grid2d_kernel<<<BLOCKS, THREADS, 0, stream>>>(pts, fp, out, n);
}